// GameOfLifeRules_89953795047597
// MI455X (gfx1250) — compile-verified
//
#include <hip/hip_runtime.h>

typedef float v2f __attribute__((ext_vector_type(2)));
typedef float v8f __attribute__((ext_vector_type(8)));

#define GOL_H 768
#define GOL_W 768

// Zero-padded weight matrix in LDS: 144 channels x 12 taps of fp32 (6912 B).
// Padding (k=9..11, ch>=nch) is zero-filled so A-fragment loads need no
// predication and are 8B-aligned ds_load_b64s.
static constexpr int WPAD_CH = 144;
static constexpr int WPAD_K  = 12;

__device__ __forceinline__ float gol_flip(float y) {
  return (y < 0.0f) ? (1.0f - y) : y;
}

// ---------------------------------------------------------------------------
// Kernel 1: per wave, and_probs for 16 consecutive pixels of one row.
//   conv(ch, pix) = sum_k W[ch,k] * X[k,pix]   via V_WMMA_F32_16X16X4_F32
//   M = 16-channel tile, N = 16 pixels, K = 12 (9 taps + 3 zero pad)
// D layout (32-bit 16x16): lane l<16 -> pixel N=l, channels M=0..7 in
// acc[0..7]; lane l+16 -> same pixel, channels M=8..15.  Product over the
// 140 channels = 8 local mults per tile + one shfl_xor(16) at the end.
// NTILES > 0: compile-time tile count (full unroll, 3*NTILES static WMMAs).
// ---------------------------------------------------------------------------
template <int NTILES>
__global__ void gol_conv_prod_kernel(const float* __restrict__ x,
                                     const float* __restrict__ wts,
                                     float* __restrict__ andp,
                                     int nch, int ntiles_rt, int nbatch) {
  __shared__ float wlds[WPAD_CH * WPAD_K];

  const int tid = threadIdx.x;
  // zero-fill, then scatter the (nch x 9) weights into the padded layout
  for (int i = tid; i < WPAD_CH * WPAD_K; i += blockDim.x) wlds[i] = 0.0f;
  __syncthreads();
  const int nw = nch * 9;
  for (int i = tid; i < nw; i += blockDim.x) {
    const int ch = i / 9, k = i % 9;
    wlds[ch * WPAD_K + k] = wts[i];
  }
  __syncthreads();

  const int wave = tid >> 5;
  const int lane = tid & 31;
  const int n    = lane & 15;   // pixel within segment (N) / channel-in-tile (M for A)
  const int half = lane >> 4;   // 0: K pair {0,1}, channels 0..7 ; 1: K pair {2,3}, ch 8..15

  const int segPerRow = GOL_W / 16;
  const int seg = blockIdx.x * (int)(blockDim.x >> 5) + wave;
  const int b   = seg / (GOL_H * segPerRow);
  if (b >= nbatch) return;
  const int rem = seg % (GOL_H * segPerRow);
  const int h   = rem / segPerRow;
  const int w0  = (rem % segPerRow) * 16;

  const float* xb = x + (size_t)b * GOL_H * GOL_W;

  // B fragments (im2col): chunk c covers K = 4c..4c+3; this lane holds
  // K = 4c + 2*half and K = 4c + 2*half + 1 for pixel column n.
  v2f bfrag[3];
#pragma unroll
  for (int c = 0; c < 3; ++c) {
#pragma unroll
    for (int j = 0; j < 2; ++j) {
      const int k = 4 * c + 2 * half + j;
      float v = 0.0f;
      if (k < 9) {
        const int dy = k / 3 - 1, dx = k % 3 - 1;
        int hh = h + dy;      hh += (hh < 0) ? GOL_H : 0; hh -= (hh >= GOL_H) ? GOL_H : 0;
        int ww = w0 + n + dx; ww += (ww < 0) ? GOL_W : 0; ww -= (ww >= GOL_W) ? GOL_W : 0;
        v = xb[hh * GOL_W + ww];
      }
      bfrag[c][j] = v;
    }
  }

  const int ntiles = (NTILES > 0) ? NTILES : ntiles_rt;
  // Channels this lane holds for tile t: t*16 + half*8 + r, r=0..7.
  // Only the last tile can spill past nch; clamp there with 'limit'.
  const int limit = nch - (ntiles - 1) * 16 - half * 8;

  float prod = 1.0f;
#pragma unroll
  for (int t = 0; t < ((NTILES > 0) ? NTILES : 1); ++t) {
    // generic fallback path loops at runtime
    for (int tt = (NTILES > 0 ? t : 0);
         tt < (NTILES > 0 ? t + 1 : ntiles_rt); ++tt) {
      const float* wp = &wlds[(tt * 16 + n) * WPAD_K + 2 * half];
      v8f acc = {0.f, 0.f, 0.f, 0.f, 0.f, 0.f, 0.f, 0.f};
#pragma unroll
      for (int c = 0; c < 3; ++c) {
        const v2f afrag = *(const v2f*)(wp + 4 * c);  // ds_load_b64, 8B aligned
        acc = __builtin_amdgcn_wmma_f32_16x16x4_f32(false, afrag, false, bfrag[c],
                                                    (short)0, acc, false, false);
      }
      if (tt == ntiles - 1) {
#pragma unroll
        for (int r = 0; r < 8; ++r) {
          const float y = gol_flip(acc[r]);
          prod *= (r < limit) ? y : 1.0f;   // pad channels contribute 1
        }
      } else {
#pragma unroll
        for (int r = 0; r < 8; ++r) prod *= gol_flip(acc[r]);
      }
    }
  }

  // combine the two channel-halves of each pixel (lane l <-> lane l+16)
  const float other = __shfl_xor(prod, 16, 32);
  const float full  = prod * other;
  if (half == 0) {
    andp[(size_t)b * GOL_H * GOL_W + (size_t)h * GOL_W + w0 + n] = full;
  }
}

// ---------------------------------------------------------------------------
// Kernel 2: one thread per (b, w):
//   or = sum_k and[b, i3[k], w] * and[b, i4[k], w]
//   out[b, h, w] = and[b, h, w] - or        (stream over h, coalesced)
// ---------------------------------------------------------------------------
__global__ void gol_or_out_kernel(const float* __restrict__ andp,
                                  const int* __restrict__ i3,
                                  const int* __restrict__ i4,
                                  float* __restrict__ out,
                                  int npairs, int nbatch) {
  const int tid = blockIdx.x * blockDim.x + threadIdx.x;
  const int total = nbatch * GOL_W;
  if (tid >= total) return;
  const int b = tid / GOL_W;
  const int w = tid % GOL_W;

  const float* ap = andp + (size_t)b * GOL_H * GOL_W;
  float orv = 0.0f;
  for (int k = 0; k < npairs; ++k) {
    orv += ap[i3[k] * GOL_W + w] * ap[i4[k] * GOL_W + w];
  }
  float* op = out + (size_t)b * GOL_H * GOL_W;
  for (int h = 0; h < GOL_H; ++h) {
    op[h * GOL_W + w] = ap[h * GOL_W + w] - orv;
  }
}

// ---------------------------------------------------------------------------
extern "C" void kernel_launch(void* const* d_in, const int* in_sizes, int n_in,
                              void* d_out, int out_size, void* d_ws, size_t ws_size,
                              hipStream_t stream) {
  const float* x   = (const float*)d_in[0];   // (B,1,768,768) fp32
  const float* wts = (const float*)d_in[1];   // (140,1,3,3)   fp32
  const int*   i3  = (const int*)d_in[2];     // (504,) int32
  const int*   i4  = (const int*)d_in[3];     // (504,) int32
  float* out = (float*)d_out;                 // (B,1,768,768) fp32

  const int nbatch = in_sizes[0] / (GOL_H * GOL_W);
  const int nch    = in_sizes[1] / 9;
  const int ntiles = (nch + 15) / 16;
  const int npairs = in_sizes[2];

  float* andp = (float*)d_ws;                 // B*768*768 fp32 scratch

  const int wavesTotal    = nbatch * GOL_H * (GOL_W / 16);
  const int blockThreads  = 256;               // 8 wave32 per block
  const int wavesPerBlock = blockThreads / 32;
  const int blocks1       = (wavesTotal + wavesPerBlock - 1) / wavesPerBlock;

  if (ntiles == 9) {   // nch == 140 (reference): fully unrolled, 27 static WMMAs
    gol_conv_prod_kernel<9><<<blocks1, blockThreads, 0, stream>>>(
        x, wts, andp, nch, ntiles, nbatch);
  } else {             // generic fallback
    gol_conv_prod_kernel<0><<<blocks1, blockThreads, 0, stream>>>(
        x, wts, andp, nch, ntiles, nbatch);
  }

  const int total2  = nbatch * GOL_W;
  const int blocks2 = (total2 + 255) / 256;
  gol_or_out_kernel<<<blocks2, 256, 0, stream>>>(andp, i3, i4, out, npairs, nbatch);
}